// ChamferLoss_35399120454049
// MI455X (gfx1250) — compile-verified
//
#include <hip/hip_runtime.h>

typedef __attribute__((ext_vector_type(2))) float v2f;
typedef __attribute__((ext_vector_type(8))) float v8f;

#define BB 8
#define CC 3
#define NN 4096
#define ROWS_PER_WAVE 32                  // 2 WMMA row tiles of 16
#define COL_CHUNK 1024                    // 64 WMMA col tiles of 16
#define NUM_CHUNKS (NN / COL_CHUNK)       // 4
#define NUM_ROWBLK (NN / ROWS_PER_WAVE)   // 128

// Raw hardware mins via inline asm: exactly one VALU op each, no
// canonicalize ops (fminf/minnum/med3 all get legalized with v_max x,x,x).
__device__ __forceinline__ float min2(float a, float b) {
  float r;
  asm("v_min_num_f32 %0, %1, %2" : "=v"(r) : "v"(a), "v"(b));
  return r;
}
__device__ __forceinline__ float min3(float a, float b, float c) {
  float r;
  asm("v_min3_num_f32 %0, %1, %2, %3" : "=v"(r) : "v"(a), "v"(b), "v"(c));
  return r;
}

// Order-preserving float <-> uint encoding so we can use integer atomicMin.
__device__ __forceinline__ unsigned fenc(float f) {
  unsigned b = __float_as_uint(f);
  return b ^ ((b >> 31) ? 0xFFFFFFFFu : 0x80000000u);
}
__device__ __forceinline__ float fdec(unsigned k) {
  unsigned b = k ^ ((k >> 31) ? 0x80000000u : 0xFFFFFFFFu);
  return __uint_as_float(b);
}

// Pack [B,C,N] cloud into two forms:
//   A-form {x, y, z, ||p||^2}       (row side; .w also feeds the C matrix)
//   B-form {-2x, -2y, -2z, ||p||^2} (column side, fully pre-scaled)
__global__ void prep_pts(const float* __restrict__ P,
                         float4* __restrict__ af, float4* __restrict__ bf) {
  int i = blockIdx.x * blockDim.x + threadIdx.x;
  if (i >= BB * NN) return;
  int b = i / NN, n = i % NN;
  const float* base = P + (size_t)b * CC * NN;
  float x = base[n], y = base[NN + n], z = base[2 * NN + n];
  float r = x * x + y * y + z * z;
  af[i] = make_float4(x, y, z, r);
  bf[i] = make_float4(-2.0f * x, -2.0f * y, -2.0f * z, r);
}

__global__ void init_keys(unsigned* __restrict__ keys, int n) {
  int i = blockIdx.x * blockDim.x + threadIdx.x;
  if (i < n) keys[i] = 0xFFFFFFFFu;
}

// For each row n of cloud A: min over all m of ||A[n]-Q[m]||^2, combined
// across column chunks via encoded atomicMin.  One wave = 32 rows x 1024 cols.
// D = A' x B' + C with A'=(x,y,z,1), B'=(-2x,-2y,-2z,||q||^2), C=||a||^2.
__global__ void __launch_bounds__(128)
chamfer_rowmin(const float4* __restrict__ AForm, const float4* __restrict__ BForm,
               unsigned* __restrict__ keyOut) {
  const int lane = threadIdx.x & 31;
  const int wid  = (int)((blockIdx.x * blockDim.x + threadIdx.x) >> 5);
  const int b      = wid / (NUM_CHUNKS * NUM_ROWBLK);
  const int rem    = wid % (NUM_CHUNKS * NUM_ROWBLK);
  const int chunk  = rem / NUM_ROWBLK;   // rowblk fastest: waves in a block
  const int rowblk = rem % NUM_ROWBLK;   // share the same B'-column stream
  const int c    = lane & 15;
  const int half = lane >> 4;
  const int n0   = rowblk * ROWS_PER_WAVE;

  const v2f*   Apair = (const v2f*)(AForm + (size_t)b * NN);  // 2 v2f / point
  const float* Aflat = (const float*)(AForm + (size_t)b * NN);
  const v2f*   Bpair = (const v2f*)(BForm + (size_t)b * NN);

  // A tiles: 16x4 f32 A layout -> lanes 0-15 hold (K0,K1)=(x,y),
  // lanes 16-31 hold (K2,K3)=(z,1).  One b64 load + one fixup per tile.
  v2f a[2];
#pragma unroll
  for (int r = 0; r < 2; ++r) {
    a[r] = Apair[(size_t)(n0 + r * 16 + c) * 2 + half];
    if (half) a[r].y = 1.0f;   // K3 of A' is the constant 1
  }

  // Loop-invariant C matrices: C[n,m] = ||a_n||^2 (constant across columns).
  // Component v / lane-half h maps to row  n0 + r*16 + h*8 + v.
  v8f cm[2];
#pragma unroll
  for (int r = 0; r < 2; ++r)
#pragma unroll
    for (int v = 0; v < 8; ++v)
      cm[r][v] = Aflat[(size_t)(n0 + r * 16 + half * 8 + v) * 4 + 3];

  v8f runmin[2];
#pragma unroll
  for (int r = 0; r < 2; ++r)
#pragma unroll
    for (int v = 0; v < 8; ++v) runmin[r][v] = __builtin_inff();

  const int m0 = chunk * COL_CHUNK;
#pragma unroll 2
  for (int j = 0; j < COL_CHUNK / 16; j += 2) {
    // Two pre-scaled B' tiles: one b64 load each, zero VALU prep.
    v2f bv0 = Bpair[(size_t)(m0 + j * 16 + c) * 2 + half];
    v2f bv1 = Bpair[(size_t)(m0 + (j + 1) * 16 + c) * 2 + half];
#pragma unroll
    for (int r = 0; r < 2; ++r) {
      // Two wmma = two 16x16 tiles of exact squared pairwise distances,
      // folded with a single 3-input min per accumulator register.
      v8f d0 = __builtin_amdgcn_wmma_f32_16x16x4_f32(
          false, a[r], false, bv0, (short)0, cm[r], false, false);
      v8f d1 = __builtin_amdgcn_wmma_f32_16x16x4_f32(
          false, a[r], false, bv1, (short)0, cm[r], false, false);
#pragma unroll
      for (int v = 0; v < 8; ++v)
        runmin[r][v] = min3(runmin[r][v], d0[v], d1[v]);
    }
  }

  // Min across the 16 lanes of each half (columns); rows stay per-VGPR/half.
#pragma unroll
  for (int r = 0; r < 2; ++r)
#pragma unroll
    for (int v = 0; v < 8; ++v) {
      float m = runmin[r][v];
#pragma unroll
      for (int s = 1; s <= 8; s <<= 1) m = min2(m, __shfl_xor(m, s, 32));
      runmin[r][v] = m;
    }

  if (c == 0) {  // lanes 0 and 16 publish rows v and v+8 of each tile
    unsigned* ko = keyOut + (size_t)b * NN;
#pragma unroll
    for (int r = 0; r < 2; ++r)
#pragma unroll
      for (int v = 0; v < 8; ++v) {
        int row = n0 + r * 16 + half * 8 + v;
        atomicMin(&ko[row], fenc(runmin[r][v]));
      }
  }
}

// mean over (d1+d2)/2 == (sum d1 + sum d2) / (2*B*N)
__global__ void final_reduce(const unsigned* __restrict__ keys,
                             float* __restrict__ out) {
  __shared__ float sm[256];
  float s = 0.0f;
  for (int i = threadIdx.x; i < 2 * BB * NN; i += 256) s += fdec(keys[i]);
  sm[threadIdx.x] = s;
  __syncthreads();
  for (int w = 128; w > 0; w >>= 1) {
    if ((int)threadIdx.x < w) sm[threadIdx.x] += sm[threadIdx.x + w];
    __syncthreads();
  }
  if (threadIdx.x == 0) out[0] = sm[0] / (2.0f * BB * NN);
}

extern "C" void kernel_launch(void* const* d_in, const int* in_sizes, int n_in,
                              void* d_out, int out_size, void* d_ws, size_t ws_size,
                              hipStream_t stream) {
  const float* X = (const float*)d_in[0];
  const float* Y = (const float*)d_in[1];

  const size_t CLOUD = (size_t)BB * NN * sizeof(float4);  // 512 KB
  char* ws = (char*)d_ws;
  float4*   XA   = (float4*)(ws + 0 * CLOUD);
  float4*   XB   = (float4*)(ws + 1 * CLOUD);
  float4*   YA   = (float4*)(ws + 2 * CLOUD);
  float4*   YB   = (float4*)(ws + 3 * CLOUD);
  unsigned* keys = (unsigned*)(ws + 4 * CLOUD);           // 2*B*N uint
  unsigned* d2k  = keys;            // d2[b,n] = min_m dist^2
  unsigned* d1k  = keys + BB * NN;  // d1[b,m] = min_n dist^2

  int total = BB * NN;
  prep_pts<<<(total + 255) / 256, 256, 0, stream>>>(X, XA, XB);
  prep_pts<<<(total + 255) / 256, 256, 0, stream>>>(Y, YA, YB);
  init_keys<<<(2 * total + 255) / 256, 256, 0, stream>>>(keys, 2 * total);

  int waves  = BB * NUM_ROWBLK * NUM_CHUNKS;  // 4096 waves per direction
  int blocks = waves * 32 / 128;
  chamfer_rowmin<<<blocks, 128, 0, stream>>>(XA, YB, d2k);  // rows = X points
  chamfer_rowmin<<<blocks, 128, 0, stream>>>(YA, XB, d1k);  // rows = Y points

  final_reduce<<<1, 256, 0, stream>>>(keys, (float*)d_out);
}